// BESNumEigen3qubitModel_16286515986968
// MI455X (gfx1250) — compile-verified
//
#include <hip/hip_runtime.h>
#include <hip/hip_bf16.h>
#include <math.h>

typedef __attribute__((ext_vector_type(2))) float v2f;
typedef __attribute__((ext_vector_type(8))) float v8f;

#define NBASIS 63
#define DIM 8
#define NCOL 128   // 8x8 complex -> 128 reals: col = (j*8+k)*2 + (0=re,1=im)

// ---------------------------------------------------------------------------
// Kernel 1: build Gell-Mann expansion table Gt[64][128] (row 63 = zero pad).
// Gt[i][(j*8+k)*2 + ri] = component of G_i at (j,k).
// ---------------------------------------------------------------------------
__global__ void build_gtable(float* __restrict__ Gt) {
    int i = threadIdx.x;            // 0..63, one row per thread
    float* r = Gt + i * NCOL;
    for (int c = 0; c < NCOL; ++c) r[c] = 0.0f;
    if (i < 28) {                   // symmetric: re(j,k)=re(k,j)=1
        int cnt = 0, jj = 0, kk = 1;
        for (int j = 0; j < DIM; ++j)
            for (int k = j + 1; k < DIM; ++k) { if (cnt == i) { jj = j; kk = k; } ++cnt; }
        r[(jj * 8 + kk) * 2] = 1.0f;
        r[(kk * 8 + jj) * 2] = 1.0f;
    } else if (i < 56) {            // antisymmetric: im(j,k)=-1, im(k,j)=+1
        int t = i - 28, cnt = 0, jj = 0, kk = 1;
        for (int j = 0; j < DIM; ++j)
            for (int k = j + 1; k < DIM; ++k) { if (cnt == t) { jj = j; kk = k; } ++cnt; }
        r[(jj * 8 + kk) * 2 + 1] = -1.0f;
        r[(kk * 8 + jj) * 2 + 1] =  1.0f;
    } else if (i < 63) {            // diagonal, l = i-55 in 1..7
        int l = i - 55;
        float f = sqrtf(2.0f / (float)(l * (l + 1)));
        for (int d = 0; d < l; ++d) r[(d * 8 + d) * 2] = f;
        r[(l * 8 + l) * 2] = -(float)l * f;
    }
}

// ---------------------------------------------------------------------------
// Kernel 2: per-row inverse norm of rho_vec[B][63]
// ---------------------------------------------------------------------------
__global__ void rnorm_kernel(const float* __restrict__ vec, float* __restrict__ rn, int B) {
    int i = blockIdx.x * blockDim.x + threadIdx.x;
    if (i >= B) return;
    const float* v = vec + (size_t)i * NBASIS;
    float s = 0.0f;
    for (int k = 0; k < NBASIS; ++k) { float x = v[k]; s += x * x; }
    rn[i] = rsqrtf(s);
}

// ---------------------------------------------------------------------------
// Kernel 3: rho_t[c][b] = sum_i vn[b][i] * Gt[i][c] + (I/8 on real diag)
// via V_WMMA_F32_16X16X4_F32. Block = 256 threads = 8 waves; wave w does
// column tile w (16 of 128 cols) for a 16-row batch tile.
// A (16x4 f32): lanes 0-15 hold M, {K0,K1}; lanes 16-31 hold M, {K2,K3}.
// B (4x16 f32): mirrored — lane holds N, {K0,K1} / {K2,K3}.
// C/D (16x16 f32): VGPR r: lanes 0-15 -> (M=r, N=lane); 16-31 -> (M=r+8).
// ---------------------------------------------------------------------------
__global__ void wmma_rho(const float* __restrict__ vec, const float* __restrict__ rn,
                         const float* __restrict__ Gt, float* __restrict__ rho_t, int B) {
    const int m0   = blockIdx.x * 16;
    const int lane = threadIdx.x & 31;
    const int n0   = (threadIdx.x >> 5) * 16;      // wave id -> column tile
    const int ln   = lane & 15;
    const int kh   = (lane >> 4) * 2;              // 0 or 2: K sub-pair

    const int m = m0 + ln;
    const float scale = rn[m];
    const float* vrow = vec + (size_t)m * NBASIS;

    v8f acc = {};
#pragma unroll
    for (int kk = 0; kk < 16; ++kk) {              // K = 64 (63 + zero pad)
        const int kb = kk * 4 + kh;
        v2f a, b;
        a.x = (kb     < NBASIS ? vrow[kb]     : 0.0f) * scale;
        a.y = (kb + 1 < NBASIS ? vrow[kb + 1] : 0.0f) * scale;
        b.x = Gt[kb * NCOL + n0 + ln];
        b.y = Gt[(kb + 1) * NCOL + n0 + ln];       // row 63 is zero pad
        acc = __builtin_amdgcn_wmma_f32_16x16x4_f32(
            /*neg_a=*/false, a, /*neg_b=*/false, b,
            /*c_mod=*/(short)0, acc, /*reuse_a=*/false, /*reuse_b=*/false);
    }

    // store transposed [col][batch]; add I/8 to real diagonal components
    const int col     = n0 + ln;
    const int rowbase = m0 + ((lane >> 4) << 3);
    const int jk = col >> 1;
    const bool diag_re = ((col & 1) == 0) && ((jk >> 3) == (jk & 7));
    const float addI = diag_re ? 0.125f : 0.0f;
#pragma unroll
    for (int r = 0; r < 8; ++r)
        rho_t[(size_t)col * B + rowbase + r] = acc[r] + addI;
}

// ---------------------------------------------------------------------------
// Kernel 4: per-lane complex Hermitian 8x8 Jacobi (eigenvalues only) in LDS.
// Variant mask m: src(j,k) = ((j&~m)|(k&m), (k&~m)|(j&m));  m=0: rho,
// m=3: intra-4x4-block transpose (PT_A up to full transpose), m=1: PT_C.
// LDS layout: X[tid + 64*idx] -> lane L always hits bank L (conflict free).
// ---------------------------------------------------------------------------
#define CE(a, b) { float _lo = fminf(a, b), _hi = fmaxf(a, b); a = _lo; b = _hi; }

__global__ void __launch_bounds__(64)
eig_loss(const float* __restrict__ rho_t, const int* __restrict__ r0p,
         const int* __restrict__ r1p, float* __restrict__ out, int B) {
    __shared__ float AR[64 * 64];
    __shared__ float AI[64 * 64];
    const int tid = threadIdx.x;
    const int e = blockIdx.x * 64 + tid;
    if (e >= B) return;                       // no barriers used: safe

    const int n0 = DIM - r0p[0];
    const int n1 = DIM - r1p[0];

    float beta0 = 0.0f, beta1 = 0.0f, loss = 0.0f;

    for (int v = 0; v < 3; ++v) {
        const int m = (v == 0) ? 0 : ((v == 1) ? 3 : 1);

        // ---- load variant (coalesced: consecutive lanes -> consecutive e) ----
#pragma unroll
        for (int j = 0; j < 8; ++j)
#pragma unroll
            for (int k = 0; k < 8; ++k) {
                const int js = (j & ~m) | (k & m);
                const int ks = (k & ~m) | (j & m);
                const int c0 = (js * 8 + ks) * 2;
                AR[tid + 64 * (j * 8 + k)] = rho_t[(size_t)c0 * B + e];
                AI[tid + 64 * (j * 8 + k)] = rho_t[(size_t)(c0 + 1) * B + e];
            }

        // ---- cyclic Jacobi sweeps ----
        for (int sweep = 0; sweep < 8; ++sweep) {
            for (int p = 0; p < 7; ++p) {
                for (int q = p + 1; q < 8; ++q) {
                    const int ipq = tid + 64 * (p * 8 + q);
                    float gr = AR[ipq], gi = AI[ipq];
                    float mag2 = gr * gr + gi * gi;
                    bool ok = mag2 > 1e-36f;
                    float mag = sqrtf(mag2);
                    float inv = ok ? 1.0f / mag : 0.0f;
                    float ur = gr * inv, ui = gi * inv;      // u = e^{i phi}
                    float app = AR[tid + 64 * (p * 9)];
                    float aqq = AR[tid + 64 * (q * 9)];
                    float tau = (aqq - app) * 0.5f * inv;
                    float t = ok ? copysignf(1.0f, tau) / (fabsf(tau) + sqrtf(1.0f + tau * tau))
                                 : 0.0f;
                    float c = 1.0f / sqrtf(1.0f + t * t);
                    float s = t * c;
                    // B = A*U  (columns p,q)
#pragma unroll
                    for (int r = 0; r < 8; ++r) {
                        const int irp = tid + 64 * (r * 8 + p);
                        const int irq = tid + 64 * (r * 8 + q);
                        float xr = AR[irp], xi = AI[irp];
                        float yr = AR[irq], yi = AI[irq];
                        AR[irp] = c * xr - s * (ur * yr + ui * yi);   // c*x - s*conj(u)*y
                        AI[irp] = c * xi - s * (ur * yi - ui * yr);
                        AR[irq] = s * (ur * xr - ui * xi) + c * yr;   // s*u*x + c*y
                        AI[irq] = s * (ur * xi + ui * xr) + c * yi;
                    }
                    // A' = U^H * B  (rows p,q)
#pragma unroll
                    for (int r = 0; r < 8; ++r) {
                        const int ipr = tid + 64 * (p * 8 + r);
                        const int iqr = tid + 64 * (q * 8 + r);
                        float xr = AR[ipr], xi = AI[ipr];
                        float yr = AR[iqr], yi = AI[iqr];
                        AR[ipr] = c * xr - s * (ur * yr - ui * yi);   // c*x - s*u*y
                        AI[ipr] = c * xi - s * (ur * yi + ui * yr);
                        AR[iqr] = s * (ur * xr + ui * xi) + c * yr;   // s*conj(u)*x + c*y
                        AI[iqr] = s * (ur * xi - ui * xr) + c * yi;
                    }
                }
            }
        }

        // ---- read diagonal ----
        float d0 = AR[tid + 64 * 0],  d1 = AR[tid + 64 * 9];
        float d2 = AR[tid + 64 * 18], d3 = AR[tid + 64 * 27];
        float d4 = AR[tid + 64 * 36], d5 = AR[tid + 64 * 45];
        float d6 = AR[tid + 64 * 54], d7 = AR[tid + 64 * 63];

        if (v == 0) {
            // full sort (Batcher odd-even merge, 19 CEs), ascending
            CE(d0, d1) CE(d2, d3) CE(d4, d5) CE(d6, d7)
            CE(d0, d2) CE(d1, d3) CE(d4, d6) CE(d5, d7)
            CE(d1, d2) CE(d5, d6)
            CE(d0, d4) CE(d1, d5) CE(d2, d6) CE(d3, d7)
            CE(d2, d4) CE(d3, d5)
            CE(d1, d2) CE(d3, d4) CE(d5, d6)
            beta0 = 1.0f / (1.0f - 8.0f * d0);     // > 0
            beta1 = 1.0f / (1.0f - 8.0f * d7);     // < 0
            float ds[8] = { d0, d1, d2, d3, d4, d5, d6, d7 };
            float Slow = 0.0f, Shigh = 0.0f;
#pragma unroll
            for (int i = 0; i < 8; ++i) {
                if (i < n0)       Slow  += ds[i];
                if (i >= 8 - n1)  Shigh += ds[i];
            }
            // eig(dm0) = beta0*(w-1/8)+1/8 (order kept);  dm1: order reversed
            float l0 = beta0 * (Slow  - (float)n0 * 0.125f) + (float)n0 * 0.125f;
            float l1 = beta1 * (Shigh - (float)n1 * 0.125f) + (float)n1 * 0.125f;
            loss = (l0 + l1) * (l0 + l1);
        } else {
            float mn = fminf(fminf(fminf(d0, d1), fminf(d2, d3)),
                             fminf(fminf(d4, d5), fminf(d6, d7)));
            float mx = fmaxf(fmaxf(fmaxf(d0, d1), fmaxf(d2, d3)),
                             fmaxf(fmaxf(d4, d5), fmaxf(d6, d7)));
            // min eig PT(dm0) from mu_min (beta0>0); min eig PT(dm1) from mu_max (beta1<0)
            float t0 = beta0 * (mn - 0.125f) + 0.125f;
            float t1 = beta1 * (mx - 0.125f) + 0.125f;
            loss += t0 * t0 + t1 * t1;
        }
    }
    out[e] = loss;
}

// ---------------------------------------------------------------------------
extern "C" void kernel_launch(void* const* d_in, const int* in_sizes, int n_in,
                              void* d_out, int out_size, void* d_ws, size_t ws_size,
                              hipStream_t stream) {
    const float* rho_vec = (const float*)d_in[0];
    const int*   rank0   = (const int*)d_in[1];
    const int*   rank1   = (const int*)d_in[2];
    const int B = in_sizes[0] / NBASIS;           // 32768

    // workspace layout (floats): Gt[64*128] | rnorm[B] | rho_t[128*B]
    float* Gt   = (float*)d_ws;
    float* rn   = Gt + 64 * NCOL;
    float* rhoT = rn + B;                         // total ~17 MB for B=32768

    build_gtable<<<1, 64, 0, stream>>>(Gt);
    rnorm_kernel<<<(B + 255) / 256, 256, 0, stream>>>(rho_vec, rn, B);
    wmma_rho<<<B / 16, 256, 0, stream>>>(rho_vec, rn, Gt, rhoT, B);
    eig_loss<<<B / 64, 64, 0, stream>>>(rhoT, rank0, rank1, (float*)d_out, B);
}